// RGCN_Model_128849019287
// MI455X (gfx1250) — compile-verified
//
#include <hip/hip_runtime.h>

// ---------------------------------------------------------------------------
// RGCN (2-layer, mean aggregation, 2 relations) for MI455X / gfx1250.
// Memory-bound problem: ~1 GB of irregular gather/scatter traffic vs ~4 GFLOP
// of dense math, so dense transforms use fp32 WMMA (V_WMMA_F32_16X16X4_F32)
// to keep full precision; aggregation uses L2-resident global fp32 atomics.
// ---------------------------------------------------------------------------

typedef __attribute__((ext_vector_type(2))) float v2f;
typedef __attribute__((ext_vector_type(8))) float v8f;

#define N_NODES 100000
#define N_EDGES 1000000
#define IN_DIM 32
#define HID_DIM 64
#define NUM_REL 2

// ---------------- edge preprocessing -------------------------------------

__global__ void count_edges_kernel(const int* __restrict__ edge_index,
                                   const int* __restrict__ edge_attr,
                                   int* __restrict__ cnt) {
  int e = blockIdx.x * blockDim.x + threadIdx.x;
  if (e >= N_EDGES) return;
  int dst = edge_index[N_EDGES + e];
  int r = edge_attr[e * 2 + 1];
  atomicAdd(&cnt[r * N_NODES + dst], 1);
}

__global__ void inv_cnt_kernel(const int* __restrict__ cnt,
                               float* __restrict__ inv) {
  int i = blockIdx.x * blockDim.x + threadIdx.x;
  if (i >= NUM_REL * N_NODES) return;
  int c = cnt[i];
  inv[i] = 1.0f / (float)(c > 0 ? c : 1);
}

// ---------------- per-relation scatter-sum of source features -------------
// One thread per (edge, 4-float chunk); consecutive threads cover one edge's
// feature row -> coalesced 128/256B gathers; sums land in L2 via atomics.

template <int F>
__global__ void scatter_kernel(const float* __restrict__ X,
                               const int* __restrict__ edge_index,
                               const int* __restrict__ edge_attr,
                               float* __restrict__ S) {
  const int C = F / 4;
  int gid = blockIdx.x * blockDim.x + threadIdx.x;
  if (gid >= N_EDGES * C) return;
  int e = gid / C;
  int ch = gid % C;
  int src = edge_index[e];
  int dst = edge_index[N_EDGES + e];
  int r = edge_attr[e * 2 + 1];
  const float4 v = *(const float4*)(X + (size_t)src * F + ch * 4);
  float* out = S + (size_t)r * N_NODES * F + (size_t)dst * F + ch * 4;
  atomicAdd(out + 0, v.x);
  atomicAdd(out + 1, v.y);
  atomicAdd(out + 2, v.z);
  atomicAdd(out + 3, v.w);
}

// ---------------- dense transform via fp32 WMMA ---------------------------
// A fragment (16x4 f32, ISA 7.12.2): lanes 0-15 hold K={0,1}, lanes 16-31
// hold K={2,3} for row M = lane&15.  B fragment (4x16): same K-half split,
// column N = lane&15.  D: lane&15 = column, rows (lane>>4)*8 .. +7 in c[0..7].

template <int K>
__device__ inline v8f mm_acc(const float* __restrict__ A, float scale,
                             const float* __restrict__ B, int col, v8f c) {
  const int half = (threadIdx.x >> 4) & 1;
#pragma unroll
  for (int t = 0; t < K / 4; ++t) {
    const int k = 4 * t + 2 * half;
    v2f a, b;
    a.x = A[k] * scale;
    a.y = A[k + 1] * scale;
    b.x = B[k * HID_DIM + col];
    b.y = B[(k + 1) * HID_DIM + col];
    c = __builtin_amdgcn_wmma_f32_16x16x4_f32(false, a, false, b,
                                              (short)0, c, false, false);
  }
  return c;
}

// H[row, :] = relu( X[row,:]@root + b + (S0[row,:]*inv0)@W[0] + (S1[row,:]*inv1)@W[1] )
// Block = 128 threads = 4 waves; wave w computes output columns [16w,16w+16)
// for the 16-node row block blockIdx.x. EXEC all-1s (WMMA requirement).
template <int K>
__global__ void __launch_bounds__(128) dense_kernel(
    const float* __restrict__ X, const float* __restrict__ S,
    const float* __restrict__ inv, const float* __restrict__ root,
    const float* __restrict__ W, const float* __restrict__ bias,
    float* __restrict__ H) {
  const int lane = threadIdx.x & 31;
  const int wave = threadIdx.x >> 5;  // 0..3 -> column tile
  const int m = lane & 15;
  const int row = blockIdx.x * 16 + m;  // node for this lane's A fragment
  const int col = wave * 16 + m;        // output column for B/D fragments

  v8f c = {};
  c = mm_acc<K>(X + (size_t)row * K, 1.0f, root, col, c);
  const float i0 = inv[row];
  const float i1 = inv[N_NODES + row];
  c = mm_acc<K>(S + (size_t)row * K, i0, W, col, c);
  c = mm_acc<K>(S + (size_t)(N_NODES + row) * K, i1, W + K * HID_DIM, col, c);

  const float bc = bias[col];
  const int half = lane >> 4;
  const int r0 = blockIdx.x * 16 + half * 8;
#pragma unroll
  for (int j = 0; j < 8; ++j) {
    float v = c[j] + bc;
    H[(size_t)(r0 + j) * HID_DIM + col] = v > 0.0f ? v : 0.0f;
  }
}

// ---------------- classifier: out = H2 @ Wc + bc (64 -> 2) ----------------

__global__ void classifier_kernel(const float* __restrict__ H,
                                  const float* __restrict__ Wc,
                                  const float* __restrict__ bc,
                                  float* __restrict__ out) {
  int n = blockIdx.x * blockDim.x + threadIdx.x;
  if (n >= N_NODES) return;
  float a0 = bc[0], a1 = bc[1];
  const float* h = H + (size_t)n * HID_DIM;
#pragma unroll
  for (int j = 0; j < HID_DIM; j += 4) {
    float4 v = *(const float4*)(h + j);
    a0 += v.x * Wc[(j + 0) * 2] + v.y * Wc[(j + 1) * 2] +
          v.z * Wc[(j + 2) * 2] + v.w * Wc[(j + 3) * 2];
    a1 += v.x * Wc[(j + 0) * 2 + 1] + v.y * Wc[(j + 1) * 2 + 1] +
          v.z * Wc[(j + 2) * 2 + 1] + v.w * Wc[(j + 3) * 2 + 1];
  }
  out[n * 2 + 0] = a0;
  out[n * 2 + 1] = a1;
}

// ---------------- launcher -------------------------------------------------

extern "C" void kernel_launch(void* const* d_in, const int* in_sizes, int n_in,
                              void* d_out, int out_size, void* d_ws,
                              size_t ws_size, hipStream_t stream) {
  const float* x     = (const float*)d_in[0];
  const int*   ei    = (const int*)d_in[1];   // [2, E]
  const int*   ea    = (const int*)d_in[2];   // [E, 2], type = col 1
  const float* W1    = (const float*)d_in[3]; // [2, 32, 64]
  const float* root1 = (const float*)d_in[4]; // [32, 64]
  const float* b1    = (const float*)d_in[5];
  const float* W2    = (const float*)d_in[6]; // [2, 64, 64]
  const float* root2 = (const float*)d_in[7]; // [64, 64]
  const float* b2    = (const float*)d_in[8];
  const float* Wc    = (const float*)d_in[9]; // [64, 2]
  const float* bc    = (const float*)d_in[10];
  float* out = (float*)d_out;

  // Workspace layout (bytes), total 129,600,000:
  char* ws = (char*)d_ws;
  int*   cnt = (int*)(ws + 0);            //  2*N int      =  0.8 MB
  float* inv = (float*)(ws + 800000);     //  2*N f32      =  0.8 MB
  float* S1  = (float*)(ws + 1600000);    //  2*N*32 f32   = 25.6 MB
  float* H1  = (float*)(ws + 27200000);   //  N*64 f32     = 25.6 MB
  float* S2  = (float*)(ws + 52800000);   //  2*N*64 f32   = 51.2 MB
  float* H2  = (float*)(ws + 104000000);  //  N*64 f32     = 25.6 MB

  // Re-zero accumulators every call (deterministic; capture-safe memsets).
  hipMemsetAsync(cnt, 0, 800000, stream);
  hipMemsetAsync(S1, 0, 25600000, stream);
  hipMemsetAsync(S2, 0, 51200000, stream);

  count_edges_kernel<<<(N_EDGES + 255) / 256, 256, 0, stream>>>(ei, ea, cnt);
  inv_cnt_kernel<<<(NUM_REL * N_NODES + 255) / 256, 256, 0, stream>>>(cnt, inv);

  // Layer 1: aggregate raw x (32-dim), then WMMA dense 32->64 + ReLU.
  scatter_kernel<IN_DIM>
      <<<(N_EDGES * (IN_DIM / 4) + 255) / 256, 256, 0, stream>>>(x, ei, ea, S1);
  dense_kernel<IN_DIM>
      <<<N_NODES / 16, 128, 0, stream>>>(x, S1, inv, root1, W1, b1, H1);

  // Layer 2: aggregate H1 (64-dim), then WMMA dense 64->64 + ReLU.
  scatter_kernel<HID_DIM>
      <<<(N_EDGES * (HID_DIM / 4) + 255) / 256, 256, 0, stream>>>(H1, ei, ea, S2);
  dense_kernel<HID_DIM>
      <<<N_NODES / 16, 128, 0, stream>>>(H1, S2, inv, root2, W2, b2, H2);

  // Classifier head.
  classifier_kernel<<<(N_NODES + 255) / 256, 256, 0, stream>>>(H2, Wc, bc, out);
}